// IDWT1D_49349174231514
// MI455X (gfx1250) — compile-verified
//
#include <hip/hip_runtime.h>

// db4 analysis filters (from the reference)
__constant__ float DEC_LO[8] = {-0.010597401785069032f,  0.0328830116668852f,
                                 0.030841381835560763f, -0.18703481171909309f,
                                -0.027983769416859854f,  0.6308807679298589f,
                                 0.7148465705529157f,    0.23037781330889653f};
__constant__ float DEC_HI[8] = {-0.23037781330889653f,   0.7148465705529157f,
                                -0.6308807679298589f,   -0.027983769416859854f,
                                 0.18703481171909309f,   0.030841381835560763f,
                                -0.0328830116668852f,   -0.010597401785069032f};

typedef __attribute__((ext_vector_type(2))) float v2f;
typedef __attribute__((ext_vector_type(8))) float v8f;

#define SSTRIDE 40   // LDS row stride (words) for level-1 slice: half-wave rows land on disjoint banks
#define TSTRIDE 25   // LDS row stride (words) for tap matrix T: 25 odd -> 16 rows on distinct banks
#define NROWS   20   // level-1 slice rows per block: 32/2 + 4 halo

// Fused 2-level inverse-DWT-style circulant contraction.
//   level1: res1[b,r,c]  = sum_j H[j]*coeff0[b,((r-j)&4095)>>1,c] + G[j]*coeff1[...]
//   level2: out [b,k,c]  = sum_j H[j]*res1 [b,((k-j)&8191)>>1,c]  + G[j]*coeff2[...]
// Level 2 is evaluated as 16x24 (taps) x 24x16 (data, N = batch*channel) f32 WMMA tiles.
__global__ __launch_bounds__(128) void idwt2_fused_kernel(
    const float* __restrict__ c0,   // [4][2048][8]
    const float* __restrict__ c1,   // [4][2048][8]
    const float* __restrict__ c2,   // [4][4096][8]
    float* __restrict__ out)        // [4][8192][8]
{
    __shared__ float Tlds[16 * TSTRIDE];     // tap matrix T[16][24], tile-position invariant
    __shared__ float S[NROWS * SSTRIDE];     // level-1 result slice, columns = (b*8+c)

    const int tid = threadIdx.x;
    const int k0  = blockIdx.x * 32;         // this block's 32 level-2 output rows

    // ---- phase 0: build tap matrix T[kr][m]; j = kr + 8 - 2*(m%12) ----
    for (int e = tid; e < 16 * 24; e += 128) {
        int kr = e / 24, m = e - kr * 24;
        int mm = (m < 12) ? m : (m - 12);
        int j  = kr + 8 - 2 * mm;
        float v = 0.0f;
        if (j >= 0 && j < 8) v = (m < 12) ? DEC_LO[j] : DEC_HI[j];
        Tlds[kr * TSTRIDE + m] = v;
    }

    // ---- phase 1: level-1 filter into LDS slice (rows rlo .. rlo+19, circular mod 4096) ----
    const int rlo = (k0 >> 1) - 4;           // may be negative for block 0 (circular wrap)
    for (int e = tid; e < NROWS * 32; e += 128) {
        int s   = e >> 5;
        int col = e & 31;
        int b   = col >> 3, c = col & 7;
        int r   = (rlo + s) & 4095;
        int par = r & 1;
        int re  = r - par;                   // even
        float acc = 0.0f;
#pragma unroll
        for (int p = 0; p < 4; ++p) {
            float hj = par ? DEC_LO[2 * p + 1] : DEC_LO[2 * p];
            float gj = par ? DEC_HI[2 * p + 1] : DEC_HI[2 * p];
            int idx  = ((re - 2 * p) & 4095) >> 1;        // [0,2048)
            int off  = b * 16384 + idx * 8 + c;
            acc += hj * c0[off] + gj * c1[off];
        }
        S[s * SSTRIDE + col] = acc;
    }
    __syncthreads();

    // ---- phase 2: level-2 via chained V_WMMA_F32_16X16X4_F32 ----
    const int lane  = tid & 31;
    const int w     = tid >> 5;
    const int tile  = w >> 1;                // 0..1 : 16-row output tile
    const int half  = w & 1;                 // 0..1 : which 16 of the 32 (b,c) columns
    const int koff  = (lane >= 16) ? 2 : 0;  // K sub-offset per A/B VGPR layout
    const int mA    = lane & 15;             // A-matrix M index (= output row kr)
    const int col   = half * 16 + (lane & 15);
    const int bb    = col >> 3, cc = col & 7;
    const int k0t   = k0 + tile * 16;
    const int ilo_t = (k0t >> 1) - 4;

    v8f acc = {};
    // K chunks 0..2: B from LDS level-1 slice (lowpass half of combined)
#pragma unroll
    for (int ch = 0; ch < 3; ++ch) {
        const int cK = ch * 4;
        v2f a, b;
        a.x = Tlds[mA * TSTRIDE + cK + koff];
        a.y = Tlds[mA * TSTRIDE + cK + koff + 1];
        int s0 = 8 * tile + cK + koff;       // local slice row
        b.x = S[s0 * SSTRIDE + col];
        b.y = S[(s0 + 1) * SSTRIDE + col];
        acc = __builtin_amdgcn_wmma_f32_16x16x4_f32(false, a, false, b,
                                                    (short)0, acc, false, false);
    }
    // K chunks 3..5: B from coeff2 (highpass half of combined), circular rows
#pragma unroll
    for (int ch = 3; ch < 6; ++ch) {
        const int cK = ch * 4;
        v2f a, b;
        a.x = Tlds[mA * TSTRIDE + cK + koff];
        a.y = Tlds[mA * TSTRIDE + cK + koff + 1];
        int mrow = (cK - 12) + koff;
        int i0 = (ilo_t + mrow) & 4095;
        int i1 = (ilo_t + mrow + 1) & 4095;
        b.x = c2[bb * 32768 + i0 * 8 + cc];
        b.y = c2[bb * 32768 + i1 * 8 + cc];
        acc = __builtin_amdgcn_wmma_f32_16x16x4_f32(false, a, false, b,
                                                    (short)0, acc, false, false);
    }

    // ---- store: D VGPR r holds out row (r or r+8) at column `col` ----
    const int krbase = (lane >= 16) ? 8 : 0;
#pragma unroll
    for (int r = 0; r < 8; ++r) {
        int k = k0t + krbase + r;
        out[bb * 65536 + k * 8 + cc] = acc[r];
    }
}

extern "C" void kernel_launch(void* const* d_in, const int* in_sizes, int n_in,
                              void* d_out, int out_size, void* d_ws, size_t ws_size,
                              hipStream_t stream) {
    (void)in_sizes; (void)n_in; (void)out_size; (void)d_ws; (void)ws_size;
    const float* c0 = (const float*)d_in[0];  // [4,2048,8]
    const float* c1 = (const float*)d_in[1];  // [4,2048,8]
    const float* c2 = (const float*)d_in[2];  // [4,4096,8]
    float* out = (float*)d_out;               // [4,8192,8]
    // 8192 output rows / 32 per block = 256 blocks; 128 threads = 4 wave32s.
    idwt2_fused_kernel<<<dim3(256), dim3(128), 0, stream>>>(c0, c1, c2, out);
}